// YoloLayer_55319178772888
// MI455X (gfx1250) — compile-verified
//
#include <hip/hip_runtime.h>
#include <hip/hip_bf16.h>

#define NB 32
#define NA 3
#define NC 80
#define NH 64
#define NW 64
#define NT 60
#define CH_STRIDE (NH * NW)      // 4096
#define NCH (NA * (5 + NC))      // 255
#define EPSF 1e-16f

typedef __attribute__((ext_vector_type(2))) float v2f;
typedef __attribute__((ext_vector_type(8))) float v8f;

// ANCHORS / stride (stride = 512/64 = 8)
__constant__ float AW9[9] = {1.25f, 2.0f,  4.125f, 3.75f,  7.75f,
                             7.375f, 14.5f, 19.5f,  46.625f};
__constant__ float AH9[9] = {1.625f, 3.75f, 2.875f, 7.625f, 5.625f,
                             14.875f, 11.25f, 24.75f, 40.75f};

struct TRec {                    // per-target full record (64 B)
  float gx, gy, gw, gh;
  int   valid, sel, m, gi, gj, cls;
  float tx, ty, tw, th, cmask;
  int   pad;
};

struct TBox {                    // compact corner table for the hot loop (32 B)
  float x1, x2, y1, y2;
  float ct;                      // 0.7 * (area + EPS)
  float valid;                   // prefix-monotone 1/0
  float pad0, pad1;
};

__device__ __forceinline__ float sigf(float x) {
  return __builtin_amdgcn_rcpf(1.0f + __expf(-x));
}
// -clog(sigmoid(x)) and -clog(1 - sigmoid(x)) via softplus (no rcp)
__device__ __forceinline__ float nlog_sig(float x) {
  return fminf(__logf(1.0f + __expf(-x)), 100.0f);
}
__device__ __forceinline__ float nlog_1msig(float x) {
  return fminf(__logf(1.0f + __expf(x)), 100.0f);
}
__device__ __forceinline__ float clogf_(float x) {
  return fmaxf(__logf(x), -100.0f);
}

// ds_swizzle_b32 group-of-32 xor mode: offset = {xor[14:10], or[9:5]=0, and[4:0]=0x1f}
#define SWZ_XOR_ADD(s, m)                                                     \
  (s) += __int_as_float(__builtin_amdgcn_ds_swizzle(__float_as_int(s),        \
                                                    (((m) << 10) | 0x1f)))

// Matrix-unit wave reduction: D = ones(16x4) x B(4x16); every row of D is the
// column-sums of B; B carries each lane's partial exactly once (bijection), so
// sum of D.vgpr0 across 32 lanes = 2 * total.  Requires EXEC all-ones.
__device__ __forceinline__ float wave_reduce(float p) {
  v2f a = {1.0f, 1.0f};
  v2f b = {p, 0.0f};
  v8f c = {};
  v8f d = __builtin_amdgcn_wmma_f32_16x16x4_f32(false, a, false, b,
                                                (short)0, c, false, false);
  float s = d[0];
  SWZ_XOR_ADD(s, 16);
  SWZ_XOR_ADD(s, 8);
  SWZ_XOR_ADD(s, 4);
  SWZ_XOR_ADD(s, 2);
  SWZ_XOR_ADD(s, 1);
  return s * 0.5f;
}

__global__ void zero_out(float* out, int n) {
  int i = blockIdx.x * blockDim.x + threadIdx.x;
  if (i < n) out[i] = 0.0f;
}

// ---- Kernel 1: per-target preprocessing -------------------------------------
__global__ __launch_bounds__(64) void prep_kernel(const float* __restrict__ tgt,
                                                  TRec* __restrict__ recs,
                                                  TBox* __restrict__ boxes) {
  int b = blockIdx.x;
  int t = threadIdx.x;
  __shared__ int nzf[NT];
  float v0 = 0.f, v1 = 0.f, v2 = 0.f, v3 = 0.f, v4 = 0.f;
  if (t < NT) {
    const float* p = tgt + (b * NT + t) * 5;
    v0 = p[0]; v1 = p[1]; v2 = p[2]; v3 = p[3]; v4 = p[4];
    nzf[t] = ((v0 + v1 + v2 + v3 + v4) != 0.0f) ? 1 : 0;
  }
  __syncthreads();
  if (t >= NT) return;
  int valid = 1;
  for (int j = 0; j <= t; ++j) valid &= nzf[j];   // cumprod prefix

  float gx = v0 * NW, gy = v1 * NH, gw = v2 * NW, gh = v3 * NH;
  int gi = (int)gx, gj = (int)gy;
  gi = min(max(gi, 0), NW - 1);
  gj = min(max(gj, 0), NH - 1);

  float best = -1.0f; int bi = 0;                 // first-max argmax
  #pragma unroll
  for (int k = 0; k < 9; ++k) {
    float inter = fminf(gw, AW9[k]) * fminf(gh, AH9[k]);
    float iou = inter * __builtin_amdgcn_rcpf(gw * gh + AW9[k] * AH9[k] - inter + EPSF);
    if (iou > best) { best = iou; bi = k; }
  }
  int mapped = (bi < NA) ? bi : -1;
  int m = (mapped >= 0) ? mapped : 0;

  TRec r;
  r.gx = gx; r.gy = gy; r.gw = gw; r.gh = gh;
  r.valid = valid;
  r.sel = (mapped >= 0 && valid) ? 1 : 0;
  r.m = m; r.gi = gi; r.gj = gj;
  r.cls = min(max((int)v4, 0), NC - 1);
  r.tx = gx - (float)gi;
  r.ty = gy - (float)gj;
  r.tw = __logf(gw / AW9[m] + EPSF);
  r.th = __logf(gh / AH9[m] + EPSF);
  r.cmask = sqrtf(2.0f - v2 * v3);
  r.pad = 0;
  recs[b * NT + t] = r;

  TBox q;
  q.x1 = gx - 0.5f * gw; q.x2 = gx + 0.5f * gw;
  q.y1 = gy - 0.5f * gh; q.y2 = gy + 0.5f * gh;
  q.ct = 0.7f * (gw * gh + EPSF);
  q.valid = valid ? 1.0f : 0.0f;
  q.pad0 = 0.0f; q.pad1 = 0.0f;
  boxes[b * NT + t] = q;
}

// ---- Kernel 2: dense noobj confidence loss (the heavy pass) -----------------
__global__ __launch_bounds__(256) void noobj_kernel(const float* __restrict__ X,
                                                    const TBox* __restrict__ boxes,
                                                    float* __restrict__ out) {
  __shared__ float wsum[8];
  int tid = threadIdx.x;
  int slice = blockIdx.x >> 4;          // 16 blocks per (b,a) slice
  int b = slice / NA, a = slice % NA;

  int cell = ((blockIdx.x & 15) << 8) + tid;   // 0..4095
  int h = cell >> 6, w = cell & 63;
  int base = ((b * NCH + a * (5 + NC)) * NH + h) * NW + w;

  float x0 = X[base];
  float x1 = X[base + CH_STRIDE];
  float x2 = X[base + 2 * CH_STRIDE];
  float x3 = X[base + 3 * CH_STRIDE];
  float x4 = X[base + 4 * CH_STRIDE];

  float px = sigf(x0) + (float)w;
  float py = sigf(x1) + (float)h;
  float pw = __expf(x2) * AW9[a];
  float ph = __expf(x3) * AH9[a];
  float ax1 = px - 0.5f * pw, ax2 = px + 0.5f * pw;
  float ay1 = py - 0.5f * ph, ay2 = py + 0.5f * ph;
  float cp = 0.7f * (pw * ph);          // 0.7 * pred area

  // iou > 0.7  <=>  1.7*inter - 0.7*parea - 0.7*(tarea+eps) > 0   (denom > 0)
  const TBox* tb = boxes + b * NT;      // block-uniform -> scalar loads
  float mx = -1.0f;
  for (int t = 0; t < NT; ++t) {
    TBox q = tb[t];
    if (q.valid == 0.0f) break;         // uniform branch (prefix-monotone)
    float iw = fmaxf(fminf(ax2, q.x2) - fmaxf(ax1, q.x1), 0.0f);
    float ih = fmaxf(fminf(ay2, q.y2) - fmaxf(ay1, q.y1), 0.0f);
    float inter = iw * ih;
    float margin = fmaf(1.7f, inter, -cp) - q.ct;
    mx = fmaxf(mx, margin);             // NaN margin -> keeps mx (not ignored)
  }

  float lp = (mx > 0.0f) ? 0.0f : nlog_1msig(x4);   // -clog(1 - pconf)

  float wsv = wave_reduce(lp);          // v_wmma_f32_16x16x4_f32 + ds_swizzle
  int lane = tid & 31, wid = tid >> 5;
  if (lane == 0) wsum[wid] = wsv;
  __syncthreads();
  if (tid == 0) {
    float tot = 0.0f;
    #pragma unroll
    for (int i = 0; i < 8; ++i) tot += wsum[i];
    atomicAdd(out, tot);
  }
}

// ---- Kernel 3: sparse obj-cell losses (box, obj-conf, 80-class BCE) ---------
__global__ __launch_bounds__(64) void obj_kernel(const float* __restrict__ X,
                                                 const TRec* __restrict__ recs,
                                                 float* __restrict__ out) {
  int b = blockIdx.x;
  int t = threadIdx.x;
  __shared__ int s_sel[NT], s_m[NT], s_gi[NT], s_gj[NT], s_cls[NT];
  if (t < NT) {
    TRec r = recs[b * NT + t];
    s_sel[t] = r.sel; s_m[t] = r.m; s_gi[t] = r.gi; s_gj[t] = r.gj; s_cls[t] = r.cls;
  }
  __syncthreads();
  if (t >= NT) return;
  if (!s_sel[t]) return;

  // last-writer-wins (matches scatter-set semantics)
  for (int t2 = t + 1; t2 < NT; ++t2)
    if (s_sel[t2] && s_m[t2] == s_m[t] && s_gi[t2] == s_gi[t] && s_gj[t2] == s_gj[t])
      return;

  // class union across duplicates -> multi-hot tcls (matches .at[flat,cls].set)
  unsigned cm[3] = {0u, 0u, 0u};
  for (int t2 = 0; t2 < NT; ++t2)
    if (s_sel[t2] && s_m[t2] == s_m[t] && s_gi[t2] == s_gi[t] && s_gj[t2] == s_gj[t])
      cm[s_cls[t2] >> 5] |= 1u << (s_cls[t2] & 31);

  TRec r = recs[b * NT + t];
  int base = ((b * NCH + r.m * (5 + NC)) * NH + r.gj) * NW + r.gi;
  float x0 = X[base];
  float x1 = X[base + CH_STRIDE];
  float x2 = X[base + 2 * CH_STRIDE];
  float x3 = X[base + 3 * CH_STRIDE];
  float x4 = X[base + 4 * CH_STRIDE];

  float l = 0.0f;
  float dx = (sigf(x0) - r.tx) * r.cmask;
  float dy = (sigf(x1) - r.ty) * r.cmask;
  float dw = (x2 - r.tw) * r.cmask;
  float dh = (x3 - r.th) * r.cmask;
  l += 0.5f * (dx * dx + dy * dy + dw * dw + dh * dh);
  l += nlog_sig(x4);                            // obj conf: -clog(pconf)

  for (int c = 0; c < NC; ++c) {
    float xc = X[base + (5 + c) * CH_STRIDE];
    bool tc = ((cm[c >> 5] >> (c & 31)) & 1u) != 0u;
    l += tc ? nlog_sig(xc) : nlog_1msig(xc);    // BCE vs multi-hot tcls
  }
  atomicAdd(out, l);
}

extern "C" void kernel_launch(void* const* d_in, const int* in_sizes, int n_in,
                              void* d_out, int out_size, void* d_ws, size_t ws_size,
                              hipStream_t stream) {
  const float* x   = (const float*)d_in[0];
  const float* tgt = (const float*)d_in[1];
  float* out = (float*)d_out;
  TRec* recs  = (TRec*)d_ws;                              // 32*60*64 B = 120 KB
  TBox* boxes = (TBox*)((char*)d_ws + NB * NT * sizeof(TRec));  // +60 KB

  zero_out<<<1, 64, 0, stream>>>(out, out_size);
  prep_kernel<<<NB, 64, 0, stream>>>(tgt, recs, boxes);
  noobj_kernel<<<(NB * NA * NH * NW) / 256, 256, 0, stream>>>(x, boxes, out);
  obj_kernel<<<NB, 64, 0, stream>>>(x, recs, out);
}